// GCG_38577396253239
// MI455X (gfx1250) — compile-verified
//
#include <hip/hip_runtime.h>
#include <cstdint>
#include <cstddef>

typedef __attribute__((ext_vector_type(2))) float v2f;
typedef __attribute__((ext_vector_type(8))) float v8f;

constexpr int B = 8, C = 256, K = 19;
constexpr int N = 128 * 128;          // H*W = 16384
constexpr int KP = 32;                // classes padded to 2 WMMA tiles
constexpr int CHUNK = 128;            // pixels per LDS tile
constexpr int LPAD = CHUNK + 2;       // row stride 130 floats = 520 B (8B-aligned rows)
constexpr int PIXSPLIT = 8;           // pixel-range split across blocks
constexpr int PIXPER = N / PIXSPLIT;  // 2048

// Monotonic float <-> uint mapping so atomicMax(uint) == float max.
__device__ __forceinline__ unsigned orderKey(float f) {
  unsigned u = __float_as_uint(f);
  return (u & 0x80000000u) ? ~u : (u | 0x80000000u);
}
__device__ __forceinline__ float unorderKey(unsigned k) {
  return (k & 0x80000000u) ? __uint_as_float(k ^ 0x80000000u)
                           : __uint_as_float(~k);
}

// ---- init: zero agg + denom, set segment-max keys to identity (=-inf) ----
__global__ void k_init(float* __restrict__ agg, unsigned* __restrict__ mkey,
                       float* __restrict__ denom) {
  int i = blockIdx.x * blockDim.x + threadIdx.x;
  if (i < B * KP * C) agg[i] = 0.0f;
  if (i < B * KP) { mkey[i] = 0u; denom[i] = 0.0f; }
}

// ---- pass 1: per-pixel argmax + max logit, atomic per-segment max ----
__global__ void k_argmax(const float* __restrict__ preds,
                         uint8_t* __restrict__ seg8, float* __restrict__ sval,
                         unsigned* __restrict__ mkey) {
  int i = blockIdx.x * blockDim.x + threadIdx.x;  // [0, B*N)
  int b = i >> 14;                                 // /N
  int n = i & (N - 1);
  const float* pp = preds + (size_t)b * K * N + n;
  float best = pp[0];
  int bi = 0;
#pragma unroll
  for (int k = 1; k < K; ++k) {
    float v = pp[(size_t)k * N];
    if (v > best) { best = v; bi = k; }            // strict >: first-max tie rule
  }
  seg8[i] = (uint8_t)bi;
  sval[i] = best;
  atomicMax(&mkey[b * KP + bi], orderKey(best));
}

// ---- pass 2: w = exp(s - m[seg]); atomic segment sum for denom ----
__global__ void k_weights(const uint8_t* __restrict__ seg8,
                          const float* __restrict__ sval,
                          const unsigned* __restrict__ mkey,
                          float* __restrict__ wbuf, float* __restrict__ denom) {
  int i = blockIdx.x * blockDim.x + threadIdx.x;
  int b = i >> 14;
  int sg = seg8[i];
  float m = unorderKey(mkey[b * KP + sg]);
  float wv = expf(sval[i] - m);
  wbuf[i] = wv;
  atomicAdd(&denom[b * KP + sg], wv);
}

// ---- pass 3: agg[b,cls,c] = sum_n weight[n] * x[b,c,n]  via fp32 WMMA ----
// Block = 4 waves (128 thr). Wave w owns channels [cg*64 + w*16, +16).
// Per 128-pixel chunk: stage X tile (coalesced) and a transposed scattered
// one-hot weight tile Pt[class][pixel] in LDS, then 32 k-steps of
// V_WMMA_F32_16X16X4_F32 per class-tile (2 tiles). All fragments are
// contiguous 8B-aligned float2 in LDS -> single ds_load_b64 each, no packing.
__global__ __launch_bounds__(128) void k_aggregate(
    const float* __restrict__ x, const uint8_t* __restrict__ seg8,
    const float* __restrict__ wbuf, const float* __restrict__ denom,
    float* __restrict__ agg) {
  __shared__ float Xs[64][LPAD];   // 64*130*4  = 33.3 KB
  __shared__ float Pt[KP][LPAD];   // 32*130*4  = 16.6 KB

  const int b    = blockIdx.z;
  const int cg   = blockIdx.y;          // channel group of 64
  const int pg   = blockIdx.x;          // pixel range
  const int tid  = threadIdx.x;
  const int wave = tid >> 5;
  const int lane = tid & 31;
  const int m    = lane & 15;           // row within A frag / col within B frag
  const int koff = (lane >> 4) << 1;    // lanes 16-31 hold K=2,3 (ISA 7.12.2)

  v8f acc0 = {};  // classes 0..15
  v8f acc1 = {};  // classes 16..31

  const int pixStart = pg * PIXPER;
  for (int p0 = pixStart; p0 < pixStart + PIXPER; p0 += CHUNK) {
    // stage 64ch x 128px of x, fully coalesced
    for (int i = tid; i < 64 * CHUNK; i += 128) {
      int ch = i >> 7;
      int px = i & (CHUNK - 1);
      Xs[ch][px] = x[(size_t)(b * C + cg * 64 + ch) * N + p0 + px];
    }
    // zero weight tile (transposed: class-major)
    for (int i = tid; i < KP * CHUNK; i += 128)
      Pt[i >> 7][i & (CHUNK - 1)] = 0.0f;
    __syncthreads();
    {  // scatter softmax weight into its class row (one pixel per thread)
      int gi = b * N + p0 + tid;
      int sg = seg8[gi];
      Pt[sg][tid] = wbuf[gi] / denom[b * KP + sg];
    }
    __syncthreads();

    const float* xrow  = &Xs[wave * 16 + m][0];
    const float* prow0 = &Pt[m][0];
    const float* prow1 = &Pt[16 + m][0];
    for (int k = 0; k < CHUNK; k += 4) {
      int p = k + koff;                               // even -> 8B aligned
      v2f a  = *(const v2f*)(xrow + p);                // A: 16ch x 4px
      v2f b0 = *(const v2f*)(prow0 + p);               // B: 4px x cls 0..15
      v2f b1 = *(const v2f*)(prow1 + p);               // B: 4px x cls 16..31
      acc0 = __builtin_amdgcn_wmma_f32_16x16x4_f32(
          false, a, false, b0, (short)0, acc0, false, false);
      acc1 = __builtin_amdgcn_wmma_f32_16x16x4_f32(
          false, a, false, b1, (short)0, acc1, false, false);
    }
    __syncthreads();
  }

  // D layout: lane<16 -> N=lane, VGPR r = row M=r; lane>=16 -> M=r+8.
#pragma unroll
  for (int r = 0; r < 8; ++r) {
    int mrow = r + ((lane >> 4) << 3);
    int chan = cg * 64 + wave * 16 + mrow;
    int ncol = lane & 15;
    atomicAdd(&agg[(size_t)(b * KP + ncol) * C + chan], acc0[r]);
    atomicAdd(&agg[(size_t)(b * KP + 16 + ncol) * C + chan], acc1[r]);
  }
}

// ---- pass 4: out[b,c,n] = agg[b, seg[b,n], c] ----
__global__ void k_broadcast(const uint8_t* __restrict__ seg8,
                            const float* __restrict__ agg,
                            float* __restrict__ out) {
  int i = blockIdx.x * blockDim.x + threadIdx.x;  // flat [b][c][n]
  int n  = i & (N - 1);
  int bc = i >> 14;
  int c  = bc & (C - 1);
  int b  = bc >> 8;
  int sg = seg8[b * N + n];
  out[i] = agg[(b * KP + sg) * C + c];
}

extern "C" void kernel_launch(void* const* d_in, const int* in_sizes, int n_in,
                              void* d_out, int out_size, void* d_ws,
                              size_t ws_size, hipStream_t stream) {
  const float* x     = (const float*)d_in[0];
  const float* preds = (const float*)d_in[1];
  float* out = (float*)d_out;

  char* ws = (char*)d_ws;
  float*    agg   = (float*)(ws);                 // 8*32*256*4 = 262144 B
  unsigned* mkey  = (unsigned*)(ws + 262144);     // 1024 B
  float*    denom = (float*)(ws + 263168);        // 1024 B
  uint8_t*  seg8  = (uint8_t*)(ws + 264192);      // 131072 B
  float*    sval  = (float*)(ws + 395264);        // 524288 B
  float*    wbuf  = (float*)(ws + 919552);        // 524288 B  (total ~1.44 MB)

  k_init<<<(B * KP * C + 255) / 256, 256, 0, stream>>>(agg, mkey, denom);
  k_argmax<<<(B * N) / 256, 256, 0, stream>>>(preds, seg8, sval, mkey);
  k_weights<<<(B * N) / 256, 256, 0, stream>>>(seg8, sval, mkey, wbuf, denom);
  dim3 g3(PIXSPLIT, C / 64, B);
  k_aggregate<<<g3, 128, 0, stream>>>(x, seg8, wbuf, denom, agg);
  k_broadcast<<<(B * C * N) / 256, 256, 0, stream>>>(seg8, agg, out);
}